// DotProductPredictor_58317065945289
// MI455X (gfx1250) — compile-verified
//
#include <hip/hip_runtime.h>

// d_in[0] = h      : float32, N_NODES x 2   (in_sizes[0] = 2*N_NODES)
// d_in[1] = edges  : int64,   N_EDGES x 2   (in_sizes[1] = 2*N_EDGES)
// d_out   = -(h[edges[:,0]] * h[edges[:,1]]) : float32, N_EDGES x 2

typedef int   v4i __attribute__((ext_vector_type(4)));
typedef float v2f __attribute__((ext_vector_type(2)));
typedef float v4f __attribute__((ext_vector_type(4)));

// Branch-free main kernel: exactly two edges per thread.
//  - 2x edge rows (16B each)   : clause-paired non-temporal b128 loads (read-once stream)
//  - 4x h gathers (8B float2)  : clause-paired RT b64 loads (800KB table stays L2-resident)
//  - 2x packed negate-multiply : v_pk_mul_f32 with neg modifiers
//  - 1x output (16B)           : non-temporal b128 store (write-once stream)
// No divergent branch -> compiler can clause both edge loads before the first
// s_wait_loadcnt, doubling loads-in-flight per wave.
__global__ __launch_bounds__(256) void edge_neg_dot_pairs_kernel(
    const float* __restrict__ h,
    const v4i*  __restrict__ edges,  // one v4i per edge: {src_lo, src_hi, dst_lo, dst_hi}
    v4f* __restrict__ out4,          // one v4f per edge pair
    int n_pairs)
{
    int i = blockIdx.x * blockDim.x + threadIdx.x;  // pair index
    if (i >= n_pairs) return;

    const v2f* __restrict__ h2 = (const v2f*)h;

    v4i ea = __builtin_nontemporal_load(&edges[2 * i]);
    v4i eb = __builtin_nontemporal_load(&edges[2 * i + 1]);

    // Node indices < 100000 -> low dword of each int64 is the index.
    v2f sa = h2[(unsigned)ea.x];
    v2f da = h2[(unsigned)ea.z];
    v2f sb = h2[(unsigned)eb.x];
    v2f db = h2[(unsigned)eb.z];

    v4f r;
    r.x = -(sa.x * da.x);
    r.y = -(sa.y * da.y);
    r.z = -(sb.x * db.x);
    r.w = -(sb.y * db.y);

    __builtin_nontemporal_store(r, &out4[i]);
}

// Host-side tail: handles the final edge when n_edges is odd (dead for 16M edges).
__global__ void edge_neg_dot_tail_kernel(
    const float* __restrict__ h,
    const v4i*  __restrict__ edges,
    v2f* __restrict__ out,
    int edge_idx)
{
    if (threadIdx.x != 0 || blockIdx.x != 0) return;
    const v2f* __restrict__ h2 = (const v2f*)h;
    v4i e = __builtin_nontemporal_load(&edges[edge_idx]);
    v2f s = h2[(unsigned)e.x];
    v2f d = h2[(unsigned)e.z];
    v2f r;
    r.x = -(s.x * d.x);
    r.y = -(s.y * d.y);
    __builtin_nontemporal_store(r, &out[edge_idx]);
}

extern "C" void kernel_launch(void* const* d_in, const int* in_sizes, int n_in,
                              void* d_out, int out_size, void* d_ws, size_t ws_size,
                              hipStream_t stream)
{
    const float* h     = (const float*)d_in[0];
    const v4i*   edges = (const v4i*)d_in[1];   // 2 x int64 per edge = 16B = one v4i
    v2f*         out   = (v2f*)d_out;

    const int n_edges = in_sizes[1] / 2;        // in_sizes[1] = flat int64 count = 2*N_EDGES
    const int n_pairs = n_edges / 2;            // full pairs, branch-free kernel

    if (n_pairs > 0) {
        const int threads = 256;                // 8 wave32 waves per block
        const int blocks  = (n_pairs + threads - 1) / threads;
        edge_neg_dot_pairs_kernel<<<blocks, threads, 0, stream>>>(
            h, edges, (v4f*)out, n_pairs);
    }
    if (n_edges & 1) {                          // odd remainder: one extra edge
        edge_neg_dot_tail_kernel<<<1, 1, 0, stream>>>(h, edges, out, n_edges - 1);
    }
}